// GiddLossConstantPi_13486197309643
// MI455X (gfx1250) — compile-verified
//
#include <hip/hip_runtime.h>
#include <float.h>

#define B_ 2
#define S_ 1024
#define V_ 32000
#define NTOK (B_ * S_)
#define MASK_ID (V_ - 1)
#define NEG_CLIPF (-1000000.0f)
#define BLOCK 256
#define NCHUNK 4
#define CHUNK (V_ / NCHUNK)   /* 8000 elements per TDM chunk */
#define NV4 (V_ / 4)          /* 8000 float4s */
#define CHUNK4 (CHUNK / 4)    /* 2000 float4s per chunk */
#define MASK_V4 (MASK_ID / 4) /* vec4 index holding MASK_ID (component .w) */

typedef __attribute__((ext_vector_type(4))) unsigned int v4u_;
typedef __attribute__((ext_vector_type(8))) int v8i_;
typedef __attribute__((ext_vector_type(4))) int v4i_;

#if __has_builtin(__builtin_amdgcn_tensor_load_to_lds)
#define HAVE_TDM 1
#else
#define HAVE_TDM 0
#endif

#if __has_builtin(__builtin_amdgcn_s_wait_tensorcnt)
#define WAIT_TENSORCNT(n) __builtin_amdgcn_s_wait_tensorcnt((short)(n))
#else
#define WAIT_TENSORCNT(n) asm volatile("s_wait_tensorcnt " #n ::: "memory")
#endif

__device__ __forceinline__ float block_reduce_max(float v, float* red) {
  const int tid = threadIdx.x;
  red[tid] = v;
  __syncthreads();
  for (int s = BLOCK / 2; s > 0; s >>= 1) {
    if (tid < s) red[tid] = fmaxf(red[tid], red[tid + s]);
    __syncthreads();
  }
  float r = red[0];
  __syncthreads();
  return r;
}

__device__ __forceinline__ float block_reduce_sum(float v, float* red) {
  const int tid = threadIdx.x;
  red[tid] = v;
  __syncthreads();
  for (int s = BLOCK / 2; s > 0; s >>= 1) {
    if (tid < s) red[tid] = red[tid] + red[tid + s];
    __syncthreads();
  }
  float r = red[0];
  __syncthreads();
  return r;
}

__global__ __launch_bounds__(BLOCK) void gidd_token_kernel(
    const float* __restrict__ logits, const int* __restrict__ input_ids,
    const int* __restrict__ z_t, const float* __restrict__ attn,
    const float* __restrict__ t_arr, const float* __restrict__ pi,
    float* __restrict__ out, float* __restrict__ part) {
  __shared__ __align__(16) float tile[V_];  // full 128KB logit row via TDM
  __shared__ float red[BLOCK];
  __shared__ float lp_z_sh;

  const int token = blockIdx.x;
  const int tid = threadIdx.x;
  const int b = token / S_;
  const float tb = t_arr[b];
  const float alpha = 1.0f - tb;
  const float beta = tb;
  const float pi0 = pi[0];  // constant-pi problem: pi[v] == pi[0]
  const int id = input_ids[token];
  const int z = z_t[token];
  const float* row = logits + (size_t)token * (size_t)V_;
  const float4* tile4 = (const float4*)tile;

#if HAVE_TDM
  // ---- Tensor Data Mover: DMA the row into LDS as 4 chunks (32KB each) ----
  if ((tid >> 5) == 0) {  // wave 0 only (TDM ignores EXEC; issue exactly once)
    const unsigned lds_base = (unsigned)(unsigned long long)(void*)&tile[0];
    const unsigned long long ga = (unsigned long long)(const void*)row;
#pragma unroll
    for (int c = 0; c < NCHUNK; ++c) {
      const unsigned long long gac = ga + (unsigned long long)(c * CHUNK * 4);
      v4u_ g0;
      g0[0] = 1u;                                      // count=1, user desc
      g0[1] = lds_base + (unsigned)(c * CHUNK * 4);    // lds_addr (bytes)
      g0[2] = (unsigned)(gac & 0xFFFFFFFFu);           // global_addr[31:0]
      g0[3] = (unsigned)((gac >> 32) & 0x01FFFFFFu) |  // global_addr[56:32]
              (2u << 30);                              // type = 2 ("image")
      v8i_ g1;
      g1[0] = (int)(2u << 16);                           // data_size = 4B
      g1[1] = (int)(((unsigned)CHUNK & 0xFFFFu) << 16);  // tensor_dim0 lo16
      g1[2] = (int)(1u << 16);                           // tensor_dim1 = 1
      g1[3] = (int)((unsigned)CHUNK << 16);              // tile_dim0 = 8000
      g1[4] = 1;                                         // tile_dim1 = 1
      g1[5] = CHUNK;                                     // tensor_dim0_stride
      g1[6] = 0;
      g1[7] = 0;
      v4i_ g2 = {0, 0, 0, 0};
      v4i_ g3 = {0, 0, 0, 0};
#if defined(__clang_major__) && (__clang_major__ >= 23)
      v8i_ gz = {0, 0, 0, 0, 0, 0, 0, 0};
      __builtin_amdgcn_tensor_load_to_lds(g0, g1, g2, g3, gz, 0);
#else
      __builtin_amdgcn_tensor_load_to_lds(g0, g1, g2, g3, 0);
#endif
    }
  }
#else
  // Fallback: cooperative copy global -> LDS
  for (int v = tid; v < V_; v += BLOCK) tile[v] = row[v];
#endif

  // ---- Pass A: row max via ds_load_b128, pipelined against TDM arrivals ----
  float m = -FLT_MAX;
#define CHUNK_MAX_PASS(c)                                           \
  do {                                                              \
    for (int i = (c)*CHUNK4 + tid; i < ((c) + 1) * CHUNK4; i += BLOCK) { \
      float4 w = tile4[i];                                          \
      if (i == MASK_V4) w.w = -FLT_MAX;                             \
      m = fmaxf(m, fmaxf(fmaxf(w.x, w.y), fmaxf(w.z, w.w)));        \
    }                                                               \
  } while (0)

#if HAVE_TDM
  if ((tid >> 5) == 0) WAIT_TENSORCNT(3);
  __syncthreads();
  CHUNK_MAX_PASS(0);
  if ((tid >> 5) == 0) WAIT_TENSORCNT(2);
  __syncthreads();
  CHUNK_MAX_PASS(1);
  if ((tid >> 5) == 0) WAIT_TENSORCNT(1);
  __syncthreads();
  CHUNK_MAX_PASS(2);
  if ((tid >> 5) == 0) WAIT_TENSORCNT(0);
  __syncthreads();
  CHUNK_MAX_PASS(3);
#else
  __syncthreads();
  CHUNK_MAX_PASS(0);
  CHUNK_MAX_PASS(1);
  CHUNK_MAX_PASS(2);
  CHUNK_MAX_PASS(3);
#endif
  const float M = block_reduce_max(m, red);

  // ---- Pass A2: softmax denominator Z (vectorized LDS reads) ----
  float sacc = 0.0f;
  for (int i = tid; i < NV4; i += BLOCK) {
    float4 w = tile4[i];
    if (i == MASK_V4) w.w = -FLT_MAX;
    sacc += __expf(w.x - M) + __expf(w.y - M);
    sacc += __expf(w.z - M) + __expf(w.w - M);
  }
  const float Z = block_reduce_sum(sacc, red);
  const float invZ = 1.0f / Z;

  // ---- Pass B: KL = sum_v q*(log q - log p), capture log p[z] ----
  const float bp = beta * pi0;
  const float lq0 = fmaxf(__logf(bp), NEG_CLIPF);
  const float q_id = alpha + bp;
  const float lq_id = fmaxf(__logf(q_id), NEG_CLIPF);
  float kl_acc = 0.0f;
  for (int i = tid; i < NV4; i += BLOCK) {
    const float4 w = tile4[i];
    const float le[4] = {w.x, w.y, w.z, w.w};
#pragma unroll
    for (int j = 0; j < 4; ++j) {
      const int v = 4 * i + j;
      float l = le[j];
      if (v == MASK_ID) l = -FLT_MAX;
      const float xh = __expf(l - M) * invZ;
      const float lp = fmaxf(__logf(fmaf(alpha, xh, bp)), NEG_CLIPF);
      const bool isid = (v == id);
      kl_acc += (isid ? q_id : bp) * ((isid ? lq_id : lq0) - lp);
      if (v == z) lp_z_sh = lp;  // exactly one thread writes
    }
  }
  const float kl = block_reduce_sum(kl_acc, red);  // barriers publish lp_z_sh

  if (tid == 0) {
    const float lq_z = (z == id) ? lq_id : lq0;
    const float lr = lq_z - lp_z_sh;
    const float is_loss = expm1f(lr) - lr;
    // weights
    const float ar = 1.0f / (1.0f - tb);
    const float piv = pi[id];
    const float w_x = piv / (1.0f - tb + tb * piv);
    const float w_nx = 1.0f / tb;
    const float ew = ((z == id) ? w_x : w_nx) * ar;
    const float lw = fminf(fmaxf(ew, 0.0f), 2.0f);
    const float tot = kl + is_loss;
    const float elbo = ew * tot;
    out[1 + token] = elbo;
    const float msk = attn[token];
    float* p = part + (size_t)token * 6;
    p[0] = lw * msk;            // sum ws*mask
    p[1] = lw * kl * msk;       // sum ws*kl*mask
    p[2] = lw * is_loss * msk;  // sum ws*is*mask
    p[3] = elbo * msk;          // sum elbo*mask
    p[4] = lw * tot;            // sum loss (unmasked)
    p[5] = msk;                 // sum mask
  }
}

__global__ __launch_bounds__(BLOCK) void gidd_finalize_kernel(
    const float* __restrict__ part, float* __restrict__ out) {
  __shared__ float red[BLOCK];
  const int tid = threadIdx.x;
  float a0 = 0, a1 = 0, a2 = 0, a3 = 0, a4 = 0, a5 = 0;
  for (int i = tid; i < NTOK; i += BLOCK) {
    const float* p = part + (size_t)i * 6;
    a0 += p[0]; a1 += p[1]; a2 += p[2];
    a3 += p[3]; a4 += p[4]; a5 += p[5];
  }
  const float s0 = block_reduce_sum(a0, red);
  const float s1 = block_reduce_sum(a1, red);
  const float s2 = block_reduce_sum(a2, red);
  const float s3 = block_reduce_sum(a3, red);
  const float s4 = block_reduce_sum(a4, red);
  const float s5 = block_reduce_sum(a5, red);
  if (tid == 0) {
    const float eps = 1.1920929e-07f;  // FLT_EPSILON
    const float dws = fmaxf(s0, eps);
    out[0] = s4 / (float)NTOK;            // loss
    out[1 + NTOK] = s1 / dws;             // kl_metric
    out[2 + NTOK] = s2 / dws;             // is_metric
    out[3 + NTOK] = s3 / fmaxf(s5, eps);  // elbo_metric
  }
}

extern "C" void kernel_launch(void* const* d_in, const int* in_sizes, int n_in,
                              void* d_out, int out_size, void* d_ws, size_t ws_size,
                              hipStream_t stream) {
  const float* logits = (const float*)d_in[0];
  const int* input_ids = (const int*)d_in[1];
  const int* z_t = (const int*)d_in[2];
  const float* attn = (const float*)d_in[3];
  const float* t = (const float*)d_in[4];
  const float* pi = (const float*)d_in[5];
  float* out = (float*)d_out;
  float* part = (float*)d_ws;  // 2048 * 6 floats = 48KB of scratch

  gidd_token_kernel<<<NTOK, BLOCK, 0, stream>>>(logits, input_ids, z_t, attn,
                                                t, pi, out, part);
  gidd_finalize_kernel<<<1, BLOCK, 0, stream>>>(part, out);
}